// UniDirectionalLSTM_31525059953340
// MI455X (gfx1250) — compile-verified
//
#include <hip/hip_runtime.h>
#include <hip/hip_bf16.h>

// CDNA5 (gfx1250) fp32 LSTM using V_WMMA_F32_16X16X4_F32.
// d_out layout: hidden_seq (64*512*512) | h_f (64*512) | c_f (64*512).
// The c_f region doubles as the running cell-state buffer (updated in place);
// hidden_seq doubles as the h_{t-1} source for the recurrent GEMM.

typedef __attribute__((ext_vector_type(2))) float v2f;
typedef __attribute__((ext_vector_type(8))) float v8f;

#define BS  64
#define SEQ 512
#define IS  512
#define HSZ 512

__device__ __forceinline__ float sigf(float v) { return 1.0f / (1.0f + expf(-v)); }

// One timestep. Grid: 32 workgroups (hidden tiles of 16). Block: 128 threads = 4 waves.
// Wave w computes gate block w (i/f/g/o) as a 64(batch) x 16(hidden) fp32 tile via
// WMMA f32 16x16x4, K = 512 (x @ W_ih^T) + 512 (h_{s-1} @ W_hh^T, skipped at s==0).
__launch_bounds__(128, 1)
__global__ void lstm_step_kernel(const float* __restrict__ x,
                                 const float* __restrict__ wih,
                                 const float* __restrict__ whh,
                                 const float* __restrict__ bih,
                                 const float* __restrict__ bhh,
                                 float* __restrict__ hseq,    // (BS, SEQ, HSZ)
                                 float* __restrict__ cstate,  // (BS, HSZ), in-place
                                 int s)
{
    __shared__ float gsm[4][BS][16];   // 4 gates x 64 batch x 16 hidden = 16 KB

    const int j0   = blockIdx.x * 16;        // hidden-slice base
    const int wave = threadIdx.x >> 5;       // gate block 0..3
    const int lane = threadIdx.x & 31;
    const int m    = lane & 15;              // M row (A) / N col (B,C,D) for this lane
    const int kk   = (lane >> 4) << 1;       // K-pair select: lanes 16-31 hold K+2,K+3
    const int mhi  = (lane >> 4) << 3;       // C/D: lanes 16-31 hold M rows +8

    // 4 independent accumulator chains (one per 16-row batch tile) -> WMMA ILP.
    v8f acc[4] = {};

    // ---- x @ W_ih^T, K = 512 ----
    const float* wrow = wih + (size_t)(wave * HSZ + j0 + m) * IS;   // B: row-major W
    for (int k = 0; k < IS; k += 4) {
        v2f bvec = *(const v2f*)(wrow + k + kk);
        #pragma unroll
        for (int mt = 0; mt < 4; ++mt) {
            const float* arow = x + (size_t)(mt * 16 + m) * (SEQ * IS) + (size_t)s * IS;
            v2f avec = *(const v2f*)(arow + k + kk);
            acc[mt] = __builtin_amdgcn_wmma_f32_16x16x4_f32(
                false, avec, false, bvec, (short)0, acc[mt], false, false);
        }
    }

    // ---- h_{s-1} @ W_hh^T, K = 512 (h_0 == 0 -> skip, uniform branch) ----
    if (s > 0) {
        const float* wrow_h = whh + (size_t)(wave * HSZ + j0 + m) * HSZ;
        for (int k = 0; k < HSZ; k += 4) {
            v2f bvec = *(const v2f*)(wrow_h + k + kk);
            #pragma unroll
            for (int mt = 0; mt < 4; ++mt) {
                const float* hrow = hseq + (size_t)(mt * 16 + m) * (SEQ * HSZ)
                                        + (size_t)(s - 1) * HSZ;
                v2f avec = *(const v2f*)(hrow + k + kk);
                acc[mt] = __builtin_amdgcn_wmma_f32_16x16x4_f32(
                    false, avec, false, bvec, (short)0, acc[mt], false, false);
            }
        }
    }

    // ---- bias + stage gate tiles to LDS (C/D layout: M = v + mhi, N = m) ----
    const float bsum = bih[wave * HSZ + j0 + m] + bhh[wave * HSZ + j0 + m];
    #pragma unroll
    for (int mt = 0; mt < 4; ++mt) {
        #pragma unroll
        for (int v = 0; v < 8; ++v) {
            gsm[wave][mt * 16 + v + mhi][m] = acc[mt][v] + bsum;
        }
    }
    __syncthreads();

    // ---- LSTM cell: 64 x 16 elements over 128 threads ----
    for (int idx = threadIdx.x; idx < BS * 16; idx += 128) {
        const int b = idx >> 4;
        const int n = idx & 15;
        const int j = j0 + n;
        const float ig = sigf (gsm[0][b][n]);
        const float fg = sigf (gsm[1][b][n]);
        const float gg = tanhf(gsm[2][b][n]);
        const float og = sigf (gsm[3][b][n]);
        const float cold = (s == 0) ? 0.0f : cstate[b * HSZ + j];
        const float cnew = fg * cold + ig * gg;
        const float hnew = og * tanhf(cnew);
        cstate[b * HSZ + j] = cnew;
        hseq[(size_t)b * (SEQ * HSZ) + (size_t)s * HSZ + j] = hnew;
    }
}

// Copy h_{SEQ-1} into the h_f output slot.
__global__ void lstm_tail_kernel(const float* __restrict__ hseq,
                                 float* __restrict__ hf)
{
    const int idx = blockIdx.x * blockDim.x + threadIdx.x;   // BS*HSZ threads
    const int b = idx / HSZ;
    const int j = idx % HSZ;
    hf[idx] = hseq[(size_t)b * (SEQ * HSZ) + (size_t)(SEQ - 1) * HSZ + j];
}

extern "C" void kernel_launch(void* const* d_in, const int* in_sizes, int n_in,
                              void* d_out, int out_size, void* d_ws, size_t ws_size,
                              hipStream_t stream)
{
    (void)in_sizes; (void)n_in; (void)out_size; (void)d_ws; (void)ws_size;

    const float* x   = (const float*)d_in[0];
    const float* wih = (const float*)d_in[1];
    const float* whh = (const float*)d_in[2];
    const float* bih = (const float*)d_in[3];
    const float* bhh = (const float*)d_in[4];

    float* out   = (float*)d_out;
    float* hseq  = out;                                    // (BS, SEQ, HSZ)
    float* hf    = out + (size_t)BS * SEQ * HSZ;           // (BS, HSZ)
    float* cf    = hf  + (size_t)BS * HSZ;                 // (BS, HSZ) = running c

    for (int s = 0; s < SEQ; ++s) {
        lstm_step_kernel<<<dim3(HSZ / 16), dim3(128), 0, stream>>>(
            x, wih, whh, bih, bhh, hseq, cf, s);
    }
    lstm_tail_kernel<<<dim3((BS * HSZ) / 256), dim3(256), 0, stream>>>(hseq, hf);
}